// WaveFrontModel_63161789055686
// MI455X (gfx1250) — compile-verified
//
#include <hip/hip_runtime.h>
#include <math.h>

#define B_ 8
#define D_ 64
#define H_ 256
#define HH_ 128
#define NRAR 5
#define ROUNDS_ 5
#define W0_ (D_*(NRAR+1))        /* 384 */
#define WFIN_ (W0_ + 6*ROUNDS_)  /* 414 */
#define WSTRIDE 416
#define WPAD 512
#define NX_ 512
#define NT_ 64
#define SIGMA_INV 100.0f
#define LN_EPS_ 1e-5f

typedef __attribute__((ext_vector_type(2))) float v2f;
typedef __attribute__((ext_vector_type(8))) float v8f;

struct EncParams {
  const float *enc_w1, *enc_b1, *ln_g, *ln_b, *enc_w2, *enc_b2;
  const float *cls_w1, *cls_b1, *cls_w2, *cls_b2;
  const float *shk_w1, *shk_b1, *shk_w2, *shk_b2;
  const float *rar_w1, *rar_b1, *rar_w2, *rar_b2;
};

__device__ __forceinline__ float geluf(float x) {
  return 0.5f * x * (1.0f + erff(x * 0.70710678118654752440f));
}
__device__ __forceinline__ float sigmoidf_(float x) { return 1.0f / (1.0f + expf(-x)); }
__device__ __forceinline__ float softplusf_(float x) {
  return fmaxf(x, 0.0f) + log1pf(expf(-fabsf(x)));
}

// All-threads block reduce-sum; result broadcast to every thread.
template <int NTHREADS>
__device__ __forceinline__ float block_sum(float v, float* red, int tid) {
  red[tid] = v;
  __syncthreads();
  for (int s = NTHREADS >> 1; s > 0; s >>= 1) {
    if (tid < s) red[tid] += red[tid + s];
    __syncthreads();
  }
  float r = red[0];
  __syncthreads();
  return r;
}

// ---------------------------------------------------------------------------
// Stage 1: h = LayerNorm(gelu(feats @ enc_w1 + enc_b1)) per row. 256 thr/row.
// ---------------------------------------------------------------------------
__global__ void k_enc_stage1(const float* __restrict__ discs,
                             const float* __restrict__ w1, const float* __restrict__ b1,
                             const float* __restrict__ ln_g, const float* __restrict__ ln_b,
                             float* __restrict__ hn) {
  int row = blockIdx.x;   // 0..511
  int j = threadIdx.x;    // 0..255
  __shared__ float red[H_];
  float f0 = discs[row * 3 + 1];
  float f1 = discs[row * 3 + 2];
  float h = f0 * w1[j] + f1 * w1[H_ + j] + b1[j];
  h = geluf(h);
  float mu = block_sum<H_>(h, red, j) * (1.0f / H_);
  float d = h - mu;
  float var = block_sum<H_>(d * d, red, j) * (1.0f / H_);
  hn[row * H_ + j] = d * rsqrtf(var + LN_EPS_) * ln_g[j] + ln_b[j];
}

// ---------------------------------------------------------------------------
// fp32 WMMA GEMM: Out[MxN] = act(A[MxK] @ W[KxN] + bias[N]) * maskRow[M]?
// One wave (32 threads) per 16x16 output tile; K-loop of V_WMMA_F32_16X16X4_F32.
// A 16x4 fp32 layout: lane = {row=lane&15, khalf=lane>>4}; a[j]=A[row][k+2*kh+j]
// B 4x16 fp32 layout: b[j]=W[k+2*kh+j][col=lane&15]
// C/D 16x16 fp32:     D[r + 8*kh][lane&15] = c[r]
// ---------------------------------------------------------------------------
__global__ void k_gemm_wmma(const float* __restrict__ A, const float* __restrict__ Wt,
                            const float* __restrict__ bias, const float* __restrict__ maskRow,
                            float* __restrict__ Out, int M, int N, int K, int act) {
  int ntn = N >> 4;
  int tile = blockIdx.x;
  int tm = tile / ntn, tn = tile % ntn;
  int lane = threadIdx.x;      // 0..31
  int row = lane & 15;
  int kh = lane >> 4;          // 0/1
  const float* Arow = A + (size_t)(tm * 16 + row) * K;
  v8f c = {};
  for (int k = 0; k < K; k += 4) {
    v2f av, bv;
    int ka = k + 2 * kh;
    av.x = Arow[ka + 0];
    av.y = Arow[ka + 1];
    bv.x = Wt[(size_t)(ka + 0) * N + tn * 16 + row];
    bv.y = Wt[(size_t)(ka + 1) * N + tn * 16 + row];
    c = __builtin_amdgcn_wmma_f32_16x16x4_f32(false, av, false, bv, (short)0, c,
                                              false, false);
  }
  int n = tn * 16 + row;
  float bs = bias[n];
#pragma unroll
  for (int r = 0; r < 8; ++r) {
    int m = tm * 16 + r + 8 * kh;
    float v = c[r] + bs;
    if (act == 1) v = geluf(v);
    else if (act == 2) v = fmaxf(v, 0.0f);
    if (maskRow) v *= maskRow[m];
    Out[(size_t)m * N + n] = v;
  }
}

// ---------------------------------------------------------------------------
// Head projections: 128 threads per row.
// ---------------------------------------------------------------------------
__global__ void k_heads(const float* __restrict__ hidC, const float* __restrict__ hidS,
                        const float* __restrict__ hidR,
                        const float* __restrict__ cls_w2, const float* __restrict__ cls_b2,
                        const float* __restrict__ shk_w2, const float* __restrict__ shk_b2,
                        const float* __restrict__ rar_w2, const float* __restrict__ rar_b2,
                        const float* __restrict__ mask,
                        float* __restrict__ is_shock, float* __restrict__ shock_speed,
                        float* __restrict__ rar01) {
  int row = blockIdx.x;
  int j = threadIdx.x;  // 0..127
  __shared__ float red[HH_];
  float hc = hidC[row * HH_ + j];
  float hs = hidS[row * HH_ + j];
  float hr = hidR[row * HH_ + j];
  float cs = block_sum<HH_>(hc * cls_w2[j], red, j);
  float ss = block_sum<HH_>(hs * shk_w2[j], red, j);
  float r0 = block_sum<HH_>(hr * rar_w2[j * 2 + 0], red, j);
  float r1 = block_sum<HH_>(hr * rar_w2[j * 2 + 1], red, j);
  if (j == 0) {
    is_shock[row] = sigmoidf_(cs + cls_b2[0]) * mask[row];
    shock_speed[row] = ss + shk_b2[0];
    rar01[row * 2 + 0] = r0 + rar_b2[0];
    rar01[row * 2 + 1] = r1 + rar_b2[1];
  }
}

// ---------------------------------------------------------------------------
// Build initial wavefront state arrays (B x W0), stride WSTRIDE.
// ---------------------------------------------------------------------------
__global__ void k_init_state(const float* __restrict__ discs, const float* __restrict__ mask,
                             const float* __restrict__ is_shock,
                             const float* __restrict__ shock_speed,
                             const float* __restrict__ rar01,
                             float* __restrict__ ox, float* __restrict__ ot,
                             float* __restrict__ sp, float* __restrict__ jm,
                             float* __restrict__ ac) {
  int g = blockIdx.x * blockDim.x + threadIdx.x;
  const int total = B_ * D_ * (NRAR + 1);
  if (g >= total) return;
  int slot = g % (NRAR + 1);
  int rowd = g / (NRAR + 1);
  int b = rowd / D_, d = rowd % D_;
  int o = b * WSTRIDE + d * (NRAR + 1) + slot;
  float x_pos = discs[rowd * 3 + 0];
  float tj = discs[rowd * 3 + 2] - discs[rowd * 3 + 1];
  float is = is_shock[rowd];
  float m = mask[rowd];
  float rw = 1.0f - is;
  float s1 = rar01[rowd * 2 + 0];
  float dl = softplusf_(rar01[rowd * 2 + 1]);
  ox[o] = x_pos;
  ot[o] = 0.0f;
  if (slot == 0) {
    sp[o] = shock_speed[rowd];
    jm[o] = tj * is;
    ac[o] = m * is;
  } else {
    float frac = (float)(slot - 1) / (float)(NRAR - 1);
    sp[o] = s1 + frac * dl;
    jm[o] = tj * rw / (float)NRAR;
    ac[o] = m * rw;
  }
}

// ---------------------------------------------------------------------------
// One collision round: 1 block per batch row, 256 threads.
// ---------------------------------------------------------------------------
__global__ void k_round(float* __restrict__ ox, float* __restrict__ ot,
                        float* __restrict__ sp, float* __restrict__ jm,
                        float* __restrict__ ac, EncParams P, int W) {
  int b = blockIdx.x;
  int tid = threadIdx.x;  // 0..255
  float* OX = ox + b * WSTRIDE;
  float* OT = ot + b * WSTRIDE;
  float* SP = sp + b * WSTRIDE;
  float* JM = jm + b * WSTRIDE;
  float* AC = ac + b * WSTRIDE;

  __shared__ float skey[WPAD];
  __shared__ int sidx[WPAD];
  __shared__ float sox[WPAD], sot_[WPAD], ssp[WPAD], sjm[WPAD], sac[WPAD], stc[WPAD];
  __shared__ float red[256];
  __shared__ int redi[256];
  __shared__ float hbuf[H_], ebuf[H_];

  // 1) sort keys: pos_ref + (1-ac)*1e6, padded with +huge
  for (int i = tid; i < WPAD; i += 256) {
    if (i < W) {
      float a = AC[i];
      skey[i] = OX[i] + SP[i] * (0.5f - OT[i]) + (1.0f - a) * 1.0e6f;
    } else {
      skey[i] = 3.0e38f;
    }
    sidx[i] = i;
  }
  __syncthreads();

  // 2) bitonic sort (key asc, idx tie-break => stable order)
  for (int k = 2; k <= WPAD; k <<= 1) {
    for (int j = k >> 1; j > 0; j >>= 1) {
      for (int i = tid; i < WPAD; i += 256) {
        int p = i ^ j;
        if (p > i) {
          bool up = ((i & k) == 0);
          float ki = skey[i], kp = skey[p];
          int ii = sidx[i], ip = sidx[p];
          bool gt = (ki > kp) || (ki == kp && ii > ip);
          if (gt == up) {
            skey[i] = kp; skey[p] = ki;
            sidx[i] = ip; sidx[p] = ii;
          }
        }
      }
      __syncthreads();
    }
  }

  // 3) gather sorted state
  for (int i = tid; i < W; i += 256) {
    int w = sidx[i];
    sox[i] = OX[w]; sot_[i] = OT[w]; ssp[i] = SP[w]; sjm[i] = JM[w]; sac[i] = AC[w];
  }
  __syncthreads();

  // 4) collision times; reuse skey for tcm
  for (int i = tid; i < WPAD; i += 256) {
    float tcm = 1.0e30f, tc = 0.0f;
    if (i < W - 1) {
      float sd = ssp[i] - ssp[i + 1];
      float pd = sox[i + 1] - sox[i] + ssp[i] * sot_[i] - ssp[i + 1] * sot_[i + 1];
      float sds = sd + ((fabsf(sd) < 1e-10f) ? 1e-10f : 0.0f);
      tc = pd / sds;
      float valid = sac[i] * sac[i + 1] * ((sd > 1e-8f) ? 1.0f : 0.0f) *
                    (((tc > 0.0f) && (tc < 1.0f)) ? 1.0f : 0.0f);
      tcm = tc + (1.0f - valid) * 1.0e10f;
    }
    skey[i] = tcm;
    stc[i] = tc;
  }
  __syncthreads();

  // 5) argmin (first-index tie-break)
  float best = 3.0e38f;
  int besti = 0x7fffffff;
  for (int i = tid; i < WPAD; i += 256) {
    float v = skey[i];
    if (v < best || (v == best && i < besti)) { best = v; besti = i; }
  }
  red[tid] = best; redi[tid] = besti;
  __syncthreads();
  for (int s = 128; s > 0; s >>= 1) {
    if (tid < s) {
      if (red[tid + s] < red[tid] ||
          (red[tid + s] == red[tid] && redi[tid + s] < redi[tid])) {
        red[tid] = red[tid + s]; redi[tid] = redi[tid + s];
      }
    }
    __syncthreads();
  }
  int ei = redi[0];
  float tcm_ei = red[0];
  __syncthreads();

  float has = (tcm_ei < 1.0e9f) ? 1.0f : 0.0f;
  float coll_t = stc[ei];
  float coll_x = sox[ei] + ssp[ei] * (coll_t - sot_[ei]);

  // 6) inclusive prefix sum of sjm*sac up to ei
  float part = 0.0f;
  for (int i = tid; i <= ei; i += 256) part += sjm[i] * sac[i];
  float rho_at_left = block_sum<256>(part, red, tid);
  float rho_L = rho_at_left - sjm[ei];
  float rho_R = rho_at_left + sjm[ei + 1];

  // 7) encode new disc (feats = rho_L, rho_R), mask = has
  {
    float h = rho_L * P.enc_w1[tid] + rho_R * P.enc_w1[H_ + tid] + P.enc_b1[tid];
    h = geluf(h);
    float mu = block_sum<256>(h, red, tid) * (1.0f / H_);
    float dv = h - mu;
    float var = block_sum<256>(dv * dv, red, tid) * (1.0f / H_);
    hbuf[tid] = dv * rsqrtf(var + LN_EPS_) * P.ln_g[tid] + P.ln_b[tid];
  }
  __syncthreads();
  {
    float e = P.enc_b2[tid];
    for (int k2 = 0; k2 < H_; ++k2) e += hbuf[k2] * P.enc_w2[k2 * H_ + tid];
    ebuf[tid] = e * has;
  }
  __syncthreads();
  float cls = 0.0f, shk = 0.0f, r0 = 0.0f, r1 = 0.0f;
  if (tid < HH_) {
    float hc = P.cls_b1[tid], hs = P.shk_b1[tid], hr = P.rar_b1[tid];
    for (int k2 = 0; k2 < H_; ++k2) {
      float ev = ebuf[k2];
      hc += ev * P.cls_w1[k2 * HH_ + tid];
      hs += ev * P.shk_w1[k2 * HH_ + tid];
      hr += ev * P.rar_w1[k2 * HH_ + tid];
    }
    hc = fmaxf(hc, 0.0f);
    hs = geluf(hs);
    hr = geluf(hr);
    cls = hc * P.cls_w2[tid];
    shk = hs * P.shk_w2[tid];
    r0 = hr * P.rar_w2[tid * 2 + 0];
    r1 = hr * P.rar_w2[tid * 2 + 1];
  }
  float clsS = block_sum<256>(cls, red, tid);
  float shkS = block_sum<256>(shk, red, tid);
  float r0S = block_sum<256>(r0, red, tid);
  float r1S = block_sum<256>(r1, red, tid);

  float nis = sigmoidf_(clsS + P.cls_b2[0]) * has;
  float nss = shkS + P.shk_b2[0];
  float s1 = r0S + P.rar_b2[0];
  float d1 = softplusf_(r1S + P.rar_b2[1]);
  float new_jump = rho_R - rho_L;
  float rw = (1.0f - nis) * has;

  // 8) deactivate winners, append 6 new entries
  if (tid == 0) {
    int oi0 = sidx[ei], oi1 = sidx[ei + 1];
    AC[oi0] *= (1.0f - has);
    AC[oi1] *= (1.0f - has);
  }
  if (tid < NRAR + 1) {
    int o = W + tid;
    OX[o] = coll_x * has;
    OT[o] = coll_t * has;
    if (tid == 0) {
      SP[o] = nss * has;
      JM[o] = new_jump * nis * has;
      AC[o] = nis * has;
    } else {
      float frac = (float)(tid - 1) / (float)(NRAR - 1);
      SP[o] = (s1 + frac * d1) * has;
      JM[o] = new_jump * rw / (float)NRAR;
      AC[o] = rw;
    }
  }
}

// ---------------------------------------------------------------------------
// Final field evaluation: 1 block per (b,t); 256 threads over NX.
// ---------------------------------------------------------------------------
__global__ void k_final(const float* __restrict__ ox, const float* __restrict__ ot,
                        const float* __restrict__ sp, const float* __restrict__ jm,
                        const float* __restrict__ ac, const float* __restrict__ discs,
                        const float* __restrict__ x_grid, const float* __restrict__ t_grid,
                        float* __restrict__ out) {
  int blk = blockIdx.x;
  int b = blk / NT_, ti = blk % NT_;
  int tid = threadIdx.x;
  __shared__ float pos[WFIN_], cf[WFIN_];
  float t = t_grid[ti];
  const float* OX = ox + b * WSTRIDE;
  const float* OT = ot + b * WSTRIDE;
  const float* SP = sp + b * WSTRIDE;
  const float* JM = jm + b * WSTRIDE;
  const float* AC = ac + b * WSTRIDE;
  for (int w = tid; w < WFIN_; w += 256) {
    float dt = t - OT[w];
    pos[w] = OX[w] + SP[w] * dt;
    cf[w] = JM[w] * AC[w] * sigmoidf_(dt * SIGMA_INV);
  }
  __syncthreads();
  float rho0 = discs[(b * D_ + 0) * 3 + 1];
  for (int x = tid; x < NX_; x += 256) {
    float xg = x_grid[x];
    float acc = 0.0f;
    for (int w = 0; w < WFIN_; ++w)
      acc += cf[w] * sigmoidf_((xg - pos[w]) * SIGMA_INV);
    out[((size_t)b * NT_ + ti) * NX_ + x] = rho0 + acc;
  }
}

// ---------------------------------------------------------------------------
extern "C" void kernel_launch(void* const* d_in, const int* in_sizes, int n_in,
                              void* d_out, int out_size, void* d_ws, size_t ws_size,
                              hipStream_t stream) {
  (void)in_sizes; (void)n_in; (void)out_size; (void)ws_size;
  const float* discs = (const float*)d_in[0];
  const float* disc_mask = (const float*)d_in[1];
  const float* x_grid = (const float*)d_in[2];
  const float* t_grid = (const float*)d_in[3];
  EncParams P;
  P.enc_w1 = (const float*)d_in[4];  P.enc_b1 = (const float*)d_in[5];
  P.ln_g   = (const float*)d_in[6];  P.ln_b   = (const float*)d_in[7];
  P.enc_w2 = (const float*)d_in[8];  P.enc_b2 = (const float*)d_in[9];
  P.cls_w1 = (const float*)d_in[10]; P.cls_b1 = (const float*)d_in[11];
  P.cls_w2 = (const float*)d_in[12]; P.cls_b2 = (const float*)d_in[13];
  P.shk_w1 = (const float*)d_in[14]; P.shk_b1 = (const float*)d_in[15];
  P.shk_w2 = (const float*)d_in[16]; P.shk_b2 = (const float*)d_in[17];
  P.rar_w1 = (const float*)d_in[18]; P.rar_b1 = (const float*)d_in[19];
  P.rar_w2 = (const float*)d_in[20]; P.rar_b2 = (const float*)d_in[21];

  const int ROWS = B_ * D_;  // 512
  float* ws = (float*)d_ws;
  float* hn          = ws;                      // 512*256
  float* emb         = hn + ROWS * H_;          // 512*256
  float* hidC        = emb + ROWS * H_;         // 512*128
  float* hidS        = hidC + ROWS * HH_;
  float* hidR        = hidS + ROWS * HH_;
  float* is_shock    = hidR + ROWS * HH_;       // 512
  float* shock_speed = is_shock + ROWS;         // 512
  float* rar01       = shock_speed + ROWS;      // 1024
  float* ox          = rar01 + 2 * ROWS;
  float* ot          = ox + B_ * WSTRIDE;
  float* sp          = ot + B_ * WSTRIDE;
  float* jm          = sp + B_ * WSTRIDE;
  float* ac          = jm + B_ * WSTRIDE;

  // Encoder
  k_enc_stage1<<<ROWS, H_, 0, stream>>>(discs, P.enc_w1, P.enc_b1, P.ln_g, P.ln_b, hn);
  // emb = (hn @ enc_w2 + b2) * mask   (M=512,N=256,K=256)
  k_gemm_wmma<<<(ROWS / 16) * (H_ / 16), 32, 0, stream>>>(
      hn, P.enc_w2, P.enc_b2, disc_mask, emb, ROWS, H_, H_, 0);
  // head hidden layers (M=512,N=128,K=256)
  k_gemm_wmma<<<(ROWS / 16) * (HH_ / 16), 32, 0, stream>>>(
      emb, P.cls_w1, P.cls_b1, nullptr, hidC, ROWS, HH_, H_, 2);
  k_gemm_wmma<<<(ROWS / 16) * (HH_ / 16), 32, 0, stream>>>(
      emb, P.shk_w1, P.shk_b1, nullptr, hidS, ROWS, HH_, H_, 1);
  k_gemm_wmma<<<(ROWS / 16) * (HH_ / 16), 32, 0, stream>>>(
      emb, P.rar_w1, P.rar_b1, nullptr, hidR, ROWS, HH_, H_, 1);
  k_heads<<<ROWS, HH_, 0, stream>>>(hidC, hidS, hidR, P.cls_w2, P.cls_b2, P.shk_w2,
                                    P.shk_b2, P.rar_w2, P.rar_b2, disc_mask, is_shock,
                                    shock_speed, rar01);
  // Initial state
  {
    int total = B_ * D_ * (NRAR + 1);
    k_init_state<<<(total + 255) / 256, 256, 0, stream>>>(
        discs, disc_mask, is_shock, shock_speed, rar01, ox, ot, sp, jm, ac);
  }
  // Collision rounds
  for (int r = 0; r < ROUNDS_; ++r) {
    int W = W0_ + 6 * r;
    k_round<<<B_, 256, 0, stream>>>(ox, ot, sp, jm, ac, P, W);
  }
  // Final field
  k_final<<<B_ * NT_, 256, 0, stream>>>(ox, ot, sp, jm, ac, discs, x_grid, t_grid,
                                        (float*)d_out);
}